// CA_SAGE_37340445671536
// MI455X (gfx1250) — compile-verified
//
#include <hip/hip_runtime.h>
#include <hip/hip_bf16.h>
#include <math.h>

// ---------------------------------------------------------------------------
// CDNA5 / gfx1250 implementation.
// Dominant cost: FC1 [64x8828]@[8828x2048] fp32 (72MB weights, HBM-bound,
// ~3.1us floor at 23.3 TB/s). Strategy: fp32 WMMA (v_wmma_f32_16x16x4_f32)
// GEMM; one wave owns a 16-wide N tile across all 4 M-tiles (M=64), reusing
// each weight fragment 4x in registers. K split into chunks writing
// deterministic partial slabs (no atomics in GEMM path). K/N are template
// constants so all strides become immediate instruction offsets; weights are
// streamed with non-temporal loads (read exactly once across the grid).
// ---------------------------------------------------------------------------

typedef __attribute__((ext_vector_type(2))) float v2f;
typedef __attribute__((ext_vector_type(8))) float v8f;

#define NN   2207
#define BB   64
#define EE   151215
#define MIDC 8
#define K1   8828      // 4 * NN
#define N1   2048
#define N2   1024
#define NCH1 8
#define NCH2 4
#define LN_EPS 1e-5f

__device__ __forceinline__ int imin(int a, int b) { return a < b ? a : b; }

// ---------------------------------------------------------------------------
__global__ void k_zero(float* __restrict__ p, size_t n) {
    size_t i = (size_t)blockIdx.x * blockDim.x + threadIdx.x;
    if (i < n) p[i] = 0.0f;
}

// One wave (32 lanes) per edge: lane handles batch b and b+32.
__global__ void k_scatter(const float* __restrict__ x, const int* __restrict__ ei,
                          const float* __restrict__ ewr,
                          float* __restrict__ agg, float* __restrict__ cnt) {
    int gid = blockIdx.x * blockDim.x + threadIdx.x;
    int e    = gid >> 5;
    int lane = gid & 31;
    if (e >= EE) return;
    int src = ei[e];
    int dst = ei[EE + e];
    float w = 1.0f / (1.0f + __expf(-ewr[e]));
    const float* xr = x + (size_t)src * BB;
    float* ar = agg + (size_t)dst * BB;
    atomicAdd(&ar[lane],      w * xr[lane]);
    atomicAdd(&ar[lane + 32], w * xr[lane + 32]);
    if (lane == 0) atomicAdd(&cnt[dst], 1.0f);
}

// h[b][n][c] = relu( 0.5*a0 * x*self_w[c] + 0.5*a1 * ((agg/cnt)*w[c] + b[c]) )
// (channel-attention gate is exactly 0.5: Linear(8,0)->Linear(0,8) => sigmoid(0))
__global__ void k_node(const float* __restrict__ x, const float* __restrict__ agg,
                       const float* __restrict__ cnt,
                       const float* __restrict__ selfw, const float* __restrict__ sw,
                       const float* __restrict__ sb, const float* __restrict__ alphas,
                       float* __restrict__ H) {
    int gid = blockIdx.x * blockDim.x + threadIdx.x;   // gid = n*BB + b (coalesced)
    if (gid >= NN * BB) return;
    int n = gid / BB;
    int b = gid % BB;
    float xv = x[gid];
    float av = agg[gid] / fmaxf(cnt[n], 1.0f);
    float a0 = 0.5f * alphas[0];
    float a1 = 0.5f * alphas[1];
    float* hp = H + ((size_t)b * NN + n) * MIDC;
#pragma unroll
    for (int c = 0; c < MIDC; ++c) {
        float h = a0 * xv * selfw[c] + a1 * (av * sw[c] + sb[c]);
        hp[c] = fmaxf(h, 0.0f);
    }
}

// LayerNorm stats per batch sample: mean/rstd over N*MID = 17656 elements.
__global__ void k_lnstats(const float* __restrict__ H, float* __restrict__ st) {
    int b = blockIdx.x;
    const float* base = H + (size_t)b * NN * MIDC;
    const int M = NN * MIDC;
    __shared__ float s1[256], s2[256];
    float a = 0.0f, q = 0.0f;
    for (int i = threadIdx.x; i < M; i += 256) {
        float v = base[i];
        a += v;
        q += v * v;
    }
    s1[threadIdx.x] = a; s2[threadIdx.x] = q;
    __syncthreads();
    for (int s = 128; s > 0; s >>= 1) {
        if (threadIdx.x < s) { s1[threadIdx.x] += s1[threadIdx.x + s];
                               s2[threadIdx.x] += s2[threadIdx.x + s]; }
        __syncthreads();
    }
    if (threadIdx.x == 0) {
        float mu  = s1[0] / (float)M;
        float var = s2[0] / (float)M - mu * mu;
        st[2 * b]     = mu;
        st[2 * b + 1] = rsqrtf(var + LN_EPS);
    }
}

// Per (b,n): LN affine -> conv1(12x8)+relu+bn1 -> conv2(4x12)+relu+bn2,
// write channel-major flattened activation A0[b][c*NN + n], K1 = 4*NN.
__global__ void k_conv(const float* __restrict__ H, const float* __restrict__ st,
                       const float* __restrict__ lng, const float* __restrict__ lnb,
                       const float* __restrict__ c1w, const float* __restrict__ c1b,
                       const float* __restrict__ g1, const float* __restrict__ b1,
                       const float* __restrict__ rm1, const float* __restrict__ rv1,
                       const float* __restrict__ c2w, const float* __restrict__ c2b,
                       const float* __restrict__ g2, const float* __restrict__ b2,
                       const float* __restrict__ rm2, const float* __restrict__ rv2,
                       float* __restrict__ A0) {
    int gid = blockIdx.x * blockDim.x + threadIdx.x;   // gid = b*NN + n
    if (gid >= NN * BB) return;
    int b = gid / NN;
    int n = gid % NN;
    float mu = st[2 * b], rstd = st[2 * b + 1];
    const float* hp = H + ((size_t)b * NN + n) * MIDC;
    float hc[MIDC];
#pragma unroll
    for (int c = 0; c < MIDC; ++c)
        hc[c] = (hp[c] - mu) * rstd * lng[n * MIDC + c] + lnb[n * MIDC + c];
    float h1[12];
#pragma unroll
    for (int o = 0; o < 12; ++o) {
        float t = c1b[o];
#pragma unroll
        for (int c = 0; c < MIDC; ++c) t += hc[c] * c1w[o * MIDC + c];
        t = fmaxf(t, 0.0f);                                 // relu then BN
        h1[o] = (t - rm1[o]) * rsqrtf(rv1[o] + LN_EPS) * g1[o] + b1[o];
    }
#pragma unroll
    for (int o = 0; o < 4; ++o) {
        float t = c2b[o];
#pragma unroll
        for (int c = 0; c < 12; ++c) t += h1[c] * c2w[o * 12 + c];
        t = fmaxf(t, 0.0f);
        t = (t - rm2[o]) * rsqrtf(rv2[o] + LN_EPS) * g2[o] + b2[o];
        A0[(size_t)b * K1 + o * NN + n] = t;                // channel-major flatten
    }
}

// ---------------------------------------------------------------------------
// fp32 WMMA GEMM: Part[chunk][64 x NnC] = A[64 x KC] * B[KC x NnC] (K-chunked).
// v_wmma_f32_16x16x4_f32 fragments:
//   A 16x4 : lane&15 = M, (lane>>4)*2 = K base, 2 VGPRs (K, K+1)
//   B 4x16 : lane&15 = N, (lane>>4)*2 = K base, 2 VGPRs
//   C 16x16: VGPR v -> M = v + 8*(lane>=16), N = lane&15
// KC/NnC compile-time => all row strides are immediate instruction offsets;
// inner loop is pointer-increment + loads + 8 WMMA (2 K-steps unrolled).
// ---------------------------------------------------------------------------
template <int KC, int NnC>
__global__ void k_wmma_gemm(const float* __restrict__ A, const float* __restrict__ Bw,
                            float* __restrict__ Part,
                            int ntiles, int nchunks, int steps_per_chunk) {
    int wave = (blockIdx.x * blockDim.x + threadIdx.x) >> 5;
    int lane = threadIdx.x & 31;
    if (wave >= ntiles * nchunks) return;        // wave-uniform: EXEC stays full
    int ntile = wave % ntiles;
    int chunk = wave / ntiles;
    int ks0 = chunk * steps_per_chunk;
    int ks1 = imin(ks0 + steps_per_chunk, KC >> 2);

    int nlo = lane & 15;
    int kq  = (lane >> 4) << 1;                  // 0 or 2

    // Per-lane streaming pointers; tile strides folded into immediate offsets.
    const float* ap = A  + (size_t)nlo * KC + (ks0 << 2) + kq;
    const float* bp = Bw + ((size_t)(ks0 << 2) + kq) * NnC + ntile * 16 + nlo;

    v8f acc[4] = {};
    int ks = ks0;
    for (; ks + 2 <= ks1; ks += 2) {
        v2f b0, b1;
        b0.x = __builtin_nontemporal_load(bp);                 // weights: read-once
        b0.y = __builtin_nontemporal_load(bp + NnC);
        b1.x = __builtin_nontemporal_load(bp + 4 * NnC);
        b1.y = __builtin_nontemporal_load(bp + 5 * NnC);
#pragma unroll
        for (int mt = 0; mt < 4; ++mt) {
            const float* ar = ap + (size_t)mt * 16 * KC;
            v2f a0; a0.x = ar[0]; a0.y = ar[1];
            acc[mt] = __builtin_amdgcn_wmma_f32_16x16x4_f32(
                false, a0, false, b0, (short)0, acc[mt], false, false);
        }
#pragma unroll
        for (int mt = 0; mt < 4; ++mt) {
            const float* ar = ap + (size_t)mt * 16 * KC + 4;
            v2f a1; a1.x = ar[0]; a1.y = ar[1];
            acc[mt] = __builtin_amdgcn_wmma_f32_16x16x4_f32(
                false, a1, false, b1, (short)0, acc[mt], false, false);
        }
        ap += 8;
        bp += (size_t)8 * NnC;
    }
    if (ks < ks1) {                                            // odd-step tail
        v2f b0;
        b0.x = __builtin_nontemporal_load(bp);
        b0.y = __builtin_nontemporal_load(bp + NnC);
#pragma unroll
        for (int mt = 0; mt < 4; ++mt) {
            const float* ar = ap + (size_t)mt * 16 * KC;
            v2f a0; a0.x = ar[0]; a0.y = ar[1];
            acc[mt] = __builtin_amdgcn_wmma_f32_16x16x4_f32(
                false, a0, false, b0, (short)0, acc[mt], false, false);
        }
    }

    int mhi = (lane >> 4) << 3;                  // +8 for lanes 16..31
    float* op = Part + (size_t)chunk * BB * NnC + (size_t)mhi * NnC + ntile * 16 + nlo;
#pragma unroll
    for (int mt = 0; mt < 4; ++mt)
#pragma unroll
        for (int v = 0; v < 8; ++v)
            op[(size_t)(mt * 16 + v) * NnC] = acc[mt][v];     // imm-offset stores
}

// out[b][n] = relu( bn( sum_chunks Part + bias ) )
__global__ void k_fc_post(const float* __restrict__ Part, const float* __restrict__ bias,
                          const float* __restrict__ g, const float* __restrict__ bb,
                          const float* __restrict__ rm, const float* __restrict__ rv,
                          float* __restrict__ out, int Nn, int nchunks) {
    int idx = blockIdx.x * blockDim.x + threadIdx.x;
    if (idx >= BB * Nn) return;
    int n = idx % Nn;
    float s = bias[n];
    for (int c = 0; c < nchunks; ++c) s += Part[(size_t)c * BB * Nn + idx];
    s = (s - rm[n]) * rsqrtf(rv[n] + LN_EPS) * g[n] + bb[n];
    out[idx] = fmaxf(s, 0.0f);
}

// Final head: logits = A2[64x1024] @ fco_w[1024x2] + fco_b, softmax -> d_out.
__global__ void k_head(const float* __restrict__ A2, const float* __restrict__ w,
                       const float* __restrict__ bias, float* __restrict__ out) {
    int b = blockIdx.x;
    __shared__ float s0[256], s1[256];
    float d0 = 0.0f, d1 = 0.0f;
    const float* a = A2 + (size_t)b * N2;
    for (int k = threadIdx.x; k < N2; k += 256) {
        float v = a[k];
        d0 += v * w[k * 2];
        d1 += v * w[k * 2 + 1];
    }
    s0[threadIdx.x] = d0; s1[threadIdx.x] = d1;
    __syncthreads();
    for (int s = 128; s > 0; s >>= 1) {
        if (threadIdx.x < s) { s0[threadIdx.x] += s0[threadIdx.x + s];
                               s1[threadIdx.x] += s1[threadIdx.x + s]; }
        __syncthreads();
    }
    if (threadIdx.x == 0) {
        float l0 = s0[0] + bias[0];
        float l1 = s1[0] + bias[1];
        float m  = fmaxf(l0, l1);
        float e0 = __expf(l0 - m);
        float e1 = __expf(l1 - m);
        float inv = 1.0f / (e0 + e1);
        out[b * 2]     = e0 * inv;
        out[b * 2 + 1] = e1 * inv;
    }
}

// ---------------------------------------------------------------------------
extern "C" void kernel_launch(void* const* d_in, const int* in_sizes, int n_in,
                              void* d_out, int out_size, void* d_ws, size_t ws_size,
                              hipStream_t stream) {
    const float* x      = (const float*)d_in[0];
    const int*   ei     = (const int*)  d_in[1];
    const float* ewr    = (const float*)d_in[2];
    const float* selfw  = (const float*)d_in[3];
    const float* sw     = (const float*)d_in[4];
    const float* sb     = (const float*)d_in[5];
    const float* alphas = (const float*)d_in[6];
    // d_in[7], d_in[8]: ca_w1/ca_w2 are size-0 (gate == 0.5 analytically)
    const float* lng    = (const float*)d_in[9];
    const float* lnb    = (const float*)d_in[10];
    const float* c1w    = (const float*)d_in[11];
    const float* c1b    = (const float*)d_in[12];
    const float* g1     = (const float*)d_in[13];
    const float* b1     = (const float*)d_in[14];
    const float* rm1    = (const float*)d_in[15];
    const float* rv1    = (const float*)d_in[16];
    const float* c2w    = (const float*)d_in[17];
    const float* c2b    = (const float*)d_in[18];
    const float* g2     = (const float*)d_in[19];
    const float* b2     = (const float*)d_in[20];
    const float* rm2    = (const float*)d_in[21];
    const float* rv2    = (const float*)d_in[22];
    const float* fc1w   = (const float*)d_in[23];
    const float* fc1b   = (const float*)d_in[24];
    const float* gf1    = (const float*)d_in[25];
    const float* bf1    = (const float*)d_in[26];
    const float* rmf1   = (const float*)d_in[27];
    const float* rvf1   = (const float*)d_in[28];
    const float* fc2w   = (const float*)d_in[29];
    const float* fc2b   = (const float*)d_in[30];
    const float* gf2    = (const float*)d_in[31];
    const float* bf2    = (const float*)d_in[32];
    const float* rmf2   = (const float*)d_in[33];
    const float* rvf2   = (const float*)d_in[34];
    const float* fcow   = (const float*)d_in[35];
    const float* fcob   = (const float*)d_in[36];
    float* out = (float*)d_out;

    // Workspace carve-up (floats, 16-elem aligned slabs), ~13.4 MB total.
    float* wsf = (float*)d_ws;
    size_t off = 0;
    auto carve = [&](size_t n) { float* r = wsf + off; off += (n + 15) & ~(size_t)15; return r; };
    float* agg = carve((size_t)NN * BB + NN);     // agg then cnt, zeroed together
    float* cnt = agg + (size_t)NN * BB;
    float* H   = carve((size_t)BB * NN * MIDC);
    float* st  = carve(2 * BB);
    float* A0  = carve((size_t)BB * K1);
    float* P1  = carve((size_t)NCH1 * BB * N1);
    float* A1  = carve((size_t)BB * N1);
    float* P2  = carve((size_t)NCH2 * BB * N2);
    float* A2  = carve((size_t)BB * N2);

    // 1) zero scatter accumulators
    {
        size_t zc = (size_t)NN * BB + NN;
        k_zero<<<(unsigned)((zc + 255) / 256), 256, 0, stream>>>(agg, zc);
    }
    // 2) edge-weighted scatter-sum (one wave per edge)
    {
        size_t th = (size_t)EE * 32;
        k_scatter<<<(unsigned)((th + 255) / 256), 256, 0, stream>>>(x, ei, ewr, agg, cnt);
    }
    // 3) SAGE combine + 0.5 CA gate + relu
    k_node<<<(NN * BB + 255) / 256, 256, 0, stream>>>(x, agg, cnt, selfw, sw, sb, alphas, H);
    // 4) LayerNorm stats per batch sample
    k_lnstats<<<BB, 256, 0, stream>>>(H, st);
    // 5) LN affine + conv stack -> flattened FC1 input A0 [64 x 8828]
    k_conv<<<(NN * BB + 255) / 256, 256, 0, stream>>>(H, st, lng, lnb, c1w, c1b,
                                                      g1, b1, rm1, rv1,
                                                      c2w, c2b, g2, b2, rm2, rv2, A0);
    // 6) FC1 via fp32 WMMA: 128 N-tiles x 8 K-chunks = 1024 waves
    {
        int ntiles = N1 / 16;
        int steps  = (K1 / 4 + NCH1 - 1) / NCH1;      // 276
        int waves  = ntiles * NCH1;                   // 1024
        k_wmma_gemm<K1, N1><<<waves * 32 / 256, 256, 0, stream>>>(A0, fc1w, P1,
                                                                  ntiles, NCH1, steps);
        k_fc_post<<<(BB * N1 + 255) / 256, 256, 0, stream>>>(P1, fc1b, gf1, bf1,
                                                             rmf1, rvf1, A1, N1, NCH1);
    }
    // 7) FC2 via fp32 WMMA: 64 N-tiles x 4 K-chunks = 256 waves
    {
        int ntiles = N2 / 16;
        int steps  = (N1 / 4) / NCH2;                 // 128
        int waves  = ntiles * NCH2;                   // 256
        k_wmma_gemm<N1, N2><<<waves * 32 / 256, 256, 0, stream>>>(A1, fc2w, P2,
                                                                  ntiles, NCH2, steps);
        k_fc_post<<<(BB * N2 + 255) / 256, 256, 0, stream>>>(P2, fc2b, gf2, bf2,
                                                             rmf2, rvf2, A2, N2, NCH2);
    }
    // 8) head + softmax
    k_head<<<BB, 256, 0, stream>>>(A2, fcow, fcob, out);
    (void)in_sizes; (void)n_in; (void)out_size; (void)ws_size;
}